// MultiViTPatchMapper_738734375555
// MI455X (gfx1250) — compile-verified
//
#include <hip/hip_runtime.h>
#include <cstdint>

// Problem constants (from the reference):
//   NS=2, B=8, 3 channels, PS=16, SR=SC=64 (src grid), DR=DC=32 (dst grid)
//   src image plane: 1024x1024 f32  -> plane stride 1<<22 bytes, row 1<<12 bytes
//   dst image plane:  512x512  f32  -> plane stride 1<<20 bytes, row 1<<11 bytes
//   DP = 1024 dst patches per batch; total jobs = B*DP = 8192
#define DP_   1024
#define JOBS  8192                 // B * DP
#define PATCH_BYTES 3072           // 3 * 16 * 16 * 4
#define WAVES_PER_BLOCK 8
#define ITERS 6                    // 192 chunks of 16B / 32 lanes

// ---------------------------------------------------------------------------
// Kernel 1: zero-init the output (reference scatters onto a zero canvas).
// 8*3*512*512 = 6,291,456 floats = 1,572,864 float4 -> 6144 blocks x 256 thr.
// ---------------------------------------------------------------------------
__global__ __launch_bounds__(256)
void zero_out_kernel(float4* __restrict__ out) {
    const unsigned i = blockIdx.x * 256u + threadIdx.x;   // exact grid, no guard
    out[i] = make_float4(0.f, 0.f, 0.f, 0.f);
}

// ---------------------------------------------------------------------------
// Kernel 2: patch gather/scatter via CDNA5 async global<->LDS DMA.
// One wave per patch job. Each lane moves one 16B chunk per iteration through
// LDS using GLOBAL_LOAD_ASYNC_TO_LDS_B128 / GLOBAL_STORE_ASYNC_FROM_LDS_B128
// (ASYNCcnt-tracked, VGPR-free data path). s_wait_asynccnt 0 between the load
// batch and the store batch because the LDS side of async ops is unordered.
// ---------------------------------------------------------------------------
__global__ __launch_bounds__(256)
void patch_move_kernel(const float* __restrict__ src,        // [2,8,3,1024,1024]
                       const int*   __restrict__ img_idx,    // [8,1024]
                       const int*   __restrict__ patch_idx,  // [8,1024]
                       const int*   __restrict__ dst_idx,    // [8,1024]
                       float*       __restrict__ out)        // [8,3,512,512]
{
    __shared__ char lds_buf[WAVES_PER_BLOCK * PATCH_BYTES];  // 24 KB / block

    const int tid  = threadIdx.x;
    const int w    = tid >> 5;          // wave within block
    const int lane = tid & 31;
    const int j    = blockIdx.x * WAVES_PER_BLOCK + w;       // 0..8191, exact

    // Gather/scatter control (wave-uniform; one coalesced dword each)
    const int si = img_idx[j];          // source image 0/1
    const int sp = patch_idx[j];        // source patch 0..4095
    const int dp = dst_idx[j];          // dest   patch 0..1023
    const int b  = j >> 10;             // batch

    const int sr = sp >> 6, sc = sp & 63;   // src patch row/col in 64x64 grid
    const int dr = dp >> 5, dc = dp & 31;   // dst patch row/col in 32x32 grid

    // Byte offset of channel-0 plane for this (si,b) in src / (b) in dst.
    const unsigned src_pl0 = (unsigned)(((si * 8 + b) * 3) << 22);
    const unsigned dst_pl0 = (unsigned)((b * 3) << 20);

    // LDS byte offset of this wave's 3KB staging buffer (addrspace(3) offset:
    // low 32 bits of the generic LDS address are the workgroup-relative offset)
    const unsigned lds_base =
        (unsigned)(uintptr_t)(&lds_buf[w * PATCH_BYTES]);

    // ---- async loads: global -> LDS ------------------------------------
#pragma unroll
    for (int it = 0; it < ITERS; ++it) {
        const int k = it * 32 + lane;            // chunk 0..191
        const int c = k >> 6;                    // channel 0..2
        const int r = (k >> 2) & 15;             // patch row 0..15
        const int q = k & 3;                     // 16B chunk within 64B row
        const unsigned soff = src_pl0 + ((unsigned)c << 22)
                            + ((unsigned)(sr * 16 + r) << 12)
                            + ((unsigned)((sc * 16 + q * 4) << 2));
        const unsigned laddr = lds_base + (unsigned)(k * 16);
        asm volatile("global_load_async_to_lds_b128 %0, %1, %2"
                     :: "v"(laddr), "v"(soff), "s"(src)
                     : "memory");
    }

    // LDS writes of async loads are not ordered w.r.t. async stores' LDS reads
    asm volatile("s_wait_asynccnt 0" ::: "memory");

    // ---- async stores: LDS -> global -----------------------------------
#pragma unroll
    for (int it = 0; it < ITERS; ++it) {
        const int k = it * 32 + lane;
        const int c = k >> 6;
        const int r = (k >> 2) & 15;
        const int q = k & 3;
        const unsigned doff = dst_pl0 + ((unsigned)c << 20)
                            + ((unsigned)(dr * 16 + r) << 11)
                            + ((unsigned)((dc * 16 + q * 4) << 2));
        const unsigned laddr = lds_base + (unsigned)(k * 16);
        asm volatile("global_store_async_from_lds_b128 %0, %1, %2"
                     :: "v"(doff), "v"(laddr), "s"(out)
                     : "memory");
    }

    // Drain before the block's LDS could be reallocated (S_ENDPGM also waits)
    asm volatile("s_wait_asynccnt 0" ::: "memory");
}

// ---------------------------------------------------------------------------
extern "C" void kernel_launch(void* const* d_in, const int* in_sizes, int n_in,
                              void* d_out, int out_size, void* d_ws, size_t ws_size,
                              hipStream_t stream) {
    (void)in_sizes; (void)n_in; (void)d_ws; (void)ws_size; (void)out_size;

    const float* src       = (const float*)d_in[0];
    const int*   img_idx   = (const int*)  d_in[1];
    const int*   patch_idx = (const int*)  d_in[2];
    const int*   dst_idx   = (const int*)  d_in[3];
    float*       out       = (float*)      d_out;

    // 1) zero canvas (reference scatters onto zeros)
    zero_out_kernel<<<6144, 256, 0, stream>>>((float4*)out);

    // 2) one wave per patch: 8192 jobs / 8 waves per 256-thread block
    patch_move_kernel<<<JOBS / WAVES_PER_BLOCK, 256, 0, stream>>>(
        src, img_idx, patch_idx, dst_idx, out);
}